// HashEncoding_74079595921577
// MI455X (gfx1250) — compile-verified
//
#include <hip/hip_runtime.h>
#include <cmath>
#include <cstdint>

// ---------------------------------------------------------------------------
// Instant-NGP style multiresolution hash encoding for MI455X (gfx1250).
//
// Mapping: one thread per (point, level) pair; tid = point*16 + level.
//  - 8 hashed float2 gathers per thread, issued as one load clause (MLP=8).
//  - Tables (64 MB total) stay L2-resident (192 MB L2); the 64 MB output is
//    streamed with non-temporal stores so it never evicts the tables.
//  - wave32: a wave covers 2 points; output stores coalesce to 256B/wave.
// ---------------------------------------------------------------------------

#define HE_TABLE_SIZE 524288u
#define HE_TABLE_MASK 0x7FFFFu
#define HE_NUM_LEVELS 16
#define HE_P1 2654435761u
#define HE_P2 805459861u

typedef float f2 __attribute__((ext_vector_type(2)));

struct ResPack { float r[HE_NUM_LEVELS]; };

__global__ __launch_bounds__(256) void HashEncoding_kernel(
    const float* __restrict__ x,       // [N,3]
    const float* __restrict__ tables,  // [16, 2^19, 2]
    float* __restrict__ out,           // [N, 32]
    int n_points,
    ResPack rp)
{
    __shared__ float sres[HE_NUM_LEVELS];
    if (threadIdx.x == 0) {
        #pragma unroll
        for (int i = 0; i < HE_NUM_LEVELS; ++i) sres[i] = rp.r[i];
    }
    __syncthreads();

    const int gid   = blockIdx.x * blockDim.x + threadIdx.x;
    const int total = n_points * HE_NUM_LEVELS;
    if (gid >= total) return;

    const int n = gid >> 4;       // point index
    const int l = gid & 15;       // level index

    // Load point (x is only 6 MB; 16 threads share each point -> cache hits).
    const float px = x[3 * n + 0];
    const float py = x[3 * n + 1];
    const float pz = x[3 * n + 2];

    // Normalize to [0, 1-1e-6]: (p + 1) * 0.5, clipped (matches reference fp32 math).
    const float cmax = 0.999999f;
    const float xn = fminf(fmaxf((px + 1.0f) * 0.5f, 0.0f), cmax);
    const float yn = fminf(fmaxf((py + 1.0f) * 0.5f, 0.0f), cmax);
    const float zn = fminf(fmaxf((pz + 1.0f) * 0.5f, 0.0f), cmax);

    const float res = sres[l];    // divergent LDS read (ds_load_b32)

    const float sx = xn * res, sy = yn * res, sz = zn * res;
    const float fx = floorf(sx), fy = floorf(sy), fz = floorf(sz);
    const float wx1 = sx - fx,  wy1 = sy - fy,  wz1 = sz - fz;
    const float wx0 = 1.0f - wx1, wy0 = 1.0f - wy1, wz0 = 1.0f - wz1;

    const uint32_t ix0 = (uint32_t)fx;
    const uint32_t iy0 = (uint32_t)fy;
    const uint32_t iz0 = (uint32_t)fz;
    const uint32_t ix1 = ix0 + 1u;
    // (c+1)*P == c*P + P with u32 wraparound -> only 2 muls total.
    const uint32_t hy0 = iy0 * HE_P1, hy1 = hy0 + HE_P1;
    const uint32_t hz0 = iz0 * HE_P2, hz1 = hz0 + HE_P2;

    // Corner order must match OFFSETS = {000,100,010,001,110,101,011,111}.
    const uint32_t i0 = (ix0 ^ hy0 ^ hz0) & HE_TABLE_MASK;
    const uint32_t i1 = (ix1 ^ hy0 ^ hz0) & HE_TABLE_MASK;
    const uint32_t i2 = (ix0 ^ hy1 ^ hz0) & HE_TABLE_MASK;
    const uint32_t i3 = (ix0 ^ hy0 ^ hz1) & HE_TABLE_MASK;
    const uint32_t i4 = (ix1 ^ hy1 ^ hz0) & HE_TABLE_MASK;
    const uint32_t i5 = (ix1 ^ hy0 ^ hz1) & HE_TABLE_MASK;
    const uint32_t i6 = (ix0 ^ hy1 ^ hz1) & HE_TABLE_MASK;
    const uint32_t i7 = (ix1 ^ hy1 ^ hz1) & HE_TABLE_MASK;

    // Per-level table base; 8 gathers issued back-to-back (global_load_b64 x8)
    // before any use -> 8 outstanding loads per thread against L2.
    const f2* __restrict__ tab =
        (const f2*)(tables) + (size_t)l * (size_t)HE_TABLE_SIZE;
    const f2 g0 = tab[i0];
    const f2 g1 = tab[i1];
    const f2 g2 = tab[i2];
    const f2 g3 = tab[i3];
    const f2 g4 = tab[i4];
    const f2 g5 = tab[i5];
    const f2 g6 = tab[i6];
    const f2 g7 = tab[i7];

    // Trilinear weights in the same corner order.
    const float w0 = wx0 * wy0 * wz0;
    const float w1 = wx1 * wy0 * wz0;
    const float w2 = wx0 * wy1 * wz0;
    const float w3 = wx0 * wy0 * wz1;
    const float w4 = wx1 * wy1 * wz0;
    const float w5 = wx1 * wy0 * wz1;
    const float w6 = wx0 * wy1 * wz1;
    const float w7 = wx1 * wy1 * wz1;

    float a0 = w0 * g0.x;            float a1 = w0 * g0.y;
    a0 = fmaf(w1, g1.x, a0);         a1 = fmaf(w1, g1.y, a1);
    a0 = fmaf(w2, g2.x, a0);         a1 = fmaf(w2, g2.y, a1);
    a0 = fmaf(w3, g3.x, a0);         a1 = fmaf(w3, g3.y, a1);
    a0 = fmaf(w4, g4.x, a0);         a1 = fmaf(w4, g4.y, a1);
    a0 = fmaf(w5, g5.x, a0);         a1 = fmaf(w5, g5.y, a1);
    a0 = fmaf(w6, g6.x, a0);         a1 = fmaf(w6, g6.y, a1);
    a0 = fmaf(w7, g7.x, a0);         a1 = fmaf(w7, g7.y, a1);

    // out[n, 2*l .. 2*l+1]; lanes of a wave cover contiguous 256 B.
    // Non-temporal store: keep the 64 MB output stream out of L2 so the
    // 64 MB of hash tables stay resident.
    f2 o; o.x = a0; o.y = a1;
    f2* dst = (f2*)(out + (size_t)n * (2 * HE_NUM_LEVELS)) + l;
    __builtin_nontemporal_store(o, dst);
}

extern "C" void kernel_launch(void* const* d_in, const int* in_sizes, int n_in,
                              void* d_out, int out_size, void* d_ws, size_t ws_size,
                              hipStream_t stream) {
    const float* x      = (const float*)d_in[0];
    const float* tables = (const float*)d_in[1];
    float* out          = (float*)d_out;

    const int n_points = in_sizes[0] / 3;

    // Replicate numpy's float64 resolution computation bit-for-bit
    // (same libm exp/log/pow/floor on the host).
    ResPack rp;
    const double b = std::exp(std::log(512.0 / 16.0) / 15.0);
    for (int l = 0; l < HE_NUM_LEVELS; ++l)
        rp.r[l] = (float)std::floor(16.0 * std::pow(b, (double)l));

    const int total = n_points * HE_NUM_LEVELS;
    const int block = 256;
    const int grid  = (total + block - 1) / block;
    HashEncoding_kernel<<<grid, block, 0, stream>>>(x, tables, out, n_points, rp);
}